// HeteroGAT_36850819400093
// MI455X (gfx1250) — compile-verified
//
#include <hip/hip_runtime.h>
#include <cstdint>

// ---------------- problem constants (match reference) ----------------
#define N_TASKS 50000
#define N_DATA  20000
#define N_DEVN  4
#define E_DT    800000
#define E_DEV   200000
#define E_TT    400000
#define HC      128     // H*C
#define CDIM    64
#define D_T     32
#define D_D     16
#define D_V     12
#define ED_DT   8
#define ED_DEV  6
#define ED_TT   10
#define OUTW    224     // 32 + 64 + 64 + 64

typedef float v2f __attribute__((ext_vector_type(2)));
typedef float v8f __attribute__((ext_vector_type(8)));

// ---------------- helpers ----------------
__device__ __forceinline__ float wred32(float v) {
    // full wave32 sum reduction
    #pragma unroll
    for (int m = 16; m >= 1; m >>= 1) v += __shfl_xor(v, m, 32);
    return v;
}
// monotonic float<->uint encoding for atomic max over signed floats
__device__ __forceinline__ unsigned encf(float x) {
    unsigned u = __float_as_uint(x);
    return (u & 0x80000000u) ? ~u : (u | 0x80000000u);
}
__device__ __forceinline__ float decf(unsigned e) {
    unsigned u = (e & 0x80000000u) ? (e & 0x7FFFFFFFu) : ~e;
    return __uint_as_float(u);
}

// ---------------- WMMA fp32 GEMM: C[m, colOff+n] = A[M,K] @ W[K,N] ----------------
// One wave per 16x16 output tile, K stepped by 4 via V_WMMA_F32_16X16X4_F32.
// M % 16 == 0, K % 4 == 0, N % 16 == 0 for all uses here.
__global__ void k_gemm_wmma(const float* __restrict__ A,
                            const float* __restrict__ W,
                            float* __restrict__ Cout,
                            int M, int K, int N, int ldc, int colOff)
{
    const int wavesPerBlk = blockDim.x >> 5;
    const int wave = blockIdx.x * wavesPerBlk + ((int)threadIdx.x >> 5);
    const int lane = threadIdx.x & 31;
    const int mtiles = M >> 4;
    const int ntiles = N >> 4;
    if (wave >= mtiles * ntiles) return;   // wave-uniform guard: EXEC all-1s for WMMA
    const int mt = wave / ntiles;
    const int nt = wave - mt * ntiles;
    const int row0 = mt << 4;
    const int col0 = nt << 4;
    const int mr = lane & 15;            // A: row-in-tile; B/D: col-in-tile
    const int kh = (lane >> 4) << 1;     // lanes 0-15 -> K{0,1}, lanes 16-31 -> K{2,3}
    v8f acc = {0.f, 0.f, 0.f, 0.f, 0.f, 0.f, 0.f, 0.f};
    for (int k0 = 0; k0 < K; k0 += 4) {
        v2f av, bv;
        const float* ap = A + (size_t)(row0 + mr) * K + (k0 + kh);
        av.x = ap[0];
        av.y = ap[1];
        const float* wp = W + (size_t)(k0 + kh) * N + (col0 + mr);
        bv.x = wp[0];
        bv.y = wp[N];
        acc = __builtin_amdgcn_wmma_f32_16x16x4_f32(false, av, false, bv,
                                                    (short)0, acc, false, false);
    }
    // D layout: lanes 0-15 hold M=vgpr, lanes 16-31 hold M=vgpr+8; N = lane&15
    const int mb = (lane >> 4) << 3;
    float* cp = Cout + (size_t)(row0 + mb) * ldc + colOff + mr;
    #pragma unroll
    for (int i = 0; i < 8; ++i) cp[(size_t)i * ldc] = acc[i];
}

// tiny scalar GEMM for the 4x12 @ 12x128 device projection
__global__ void k_small_gemm(const float* __restrict__ A, const float* __restrict__ W,
                             float* __restrict__ Cout, int M, int K, int N)
{
    int idx = blockIdx.x * blockDim.x + threadIdx.x;
    if (idx >= M * N) return;
    int m = idx / N, n = idx - m * N;
    float t = 0.f;
    for (int k = 0; k < K; ++k) t += A[m * K + k] * W[k * N + n];
    Cout[idx] = t;
}

// ---------------- fills ----------------
__global__ void k_fill_f32(float* p, float v, long long n) {
    long long i = (long long)blockIdx.x * blockDim.x + threadIdx.x;
    if (i < n) p[i] = v;
}
__global__ void k_fill_u32(unsigned* p, unsigned v, long long n) {
    long long i = (long long)blockIdx.x * blockDim.x + threadIdx.x;
    if (i < n) p[i] = v;
}

// ---------------- attention prep ----------------
// wdv[k*2+h] = sum_c Wd[k,h*64+c]*ad[h*64+c]   (threads 0..63)
// wev[k*2+h] = sum_c We[k,h*64+c]*ae[h*64+c]   (threads 64..127)
__global__ void k_vec_prep(const float* __restrict__ Wd, const float* __restrict__ ad,
                           const float* __restrict__ We, const float* __restrict__ ae,
                           float* __restrict__ wdv, float* __restrict__ wev,
                           int Ddst, int edim)
{
    int t = threadIdx.x;
    if (t < 64) {
        int k = t >> 1, h = t & 1;
        if (k < Ddst) {
            float s = 0.f;
            for (int c = 0; c < CDIM; ++c) s += Wd[(size_t)k * HC + h * CDIM + c] * ad[h * CDIM + c];
            wdv[k * 2 + h] = s;
        }
    } else if (t < 128) {
        int tt = t - 64, k = tt >> 1, h = tt & 1;
        if (k < edim) {
            float s = 0.f;
            for (int c = 0; c < CDIM; ++c) s += We[(size_t)k * HC + h * CDIM + c] * ae[h * CDIM + c];
            wev[k * 2 + h] = s;
        }
    }
}

// s_src[n,h] = sum_c hs[n,h*64+c]*as[h*64+c]  — one wave per node, 2 ch/lane/head
__global__ void k_s_src(const float* __restrict__ hs, const float* __restrict__ asf,
                        float* __restrict__ s, int N)
{
    int wave = blockIdx.x * (blockDim.x >> 5) + ((int)threadIdx.x >> 5);
    int lane = threadIdx.x & 31;
    if (wave >= N) return;
    const float* b = hs + (size_t)wave * HC;
    float a0 = b[lane] * asf[lane] + b[32 + lane] * asf[32 + lane];
    float a1 = b[64 + lane] * asf[64 + lane] + b[96 + lane] * asf[96 + lane];
    a0 = wred32(a0);
    a1 = wred32(a1);
    if (lane == 0) { s[wave * 2 + 0] = a0; s[wave * 2 + 1] = a1; }
}

// s_dst[n,h] = x[n,:] . wdv[:,h]   — one thread per node
__global__ void k_s_dst(const float* __restrict__ x, const float* __restrict__ wdv,
                        float* __restrict__ s, int N, int D)
{
    int n = blockIdx.x * blockDim.x + threadIdx.x;
    if (n >= N) return;
    float t0 = 0.f, t1 = 0.f;
    const float* xr = x + (size_t)n * D;
    for (int k = 0; k < D; ++k) { t0 += xr[k] * wdv[k * 2]; t1 += xr[k] * wdv[k * 2 + 1]; }
    s[n * 2 + 0] = t0; s[n * 2 + 1] = t1;
}

// ---------------- self-loop attr (conv3): mean of incoming edge_attr ----------------
__global__ void k_loop_stats(const int* __restrict__ dst, const float* __restrict__ ea,
                             float* __restrict__ cnt, float* __restrict__ la, int E)
{
    int e = blockIdx.x * blockDim.x + threadIdx.x;
    if (e >= E) return;
    int d = dst[e];
    atomicAdd(&cnt[d], 1.f);
    for (int k = 0; k < ED_TT; ++k) atomicAdd(&la[(size_t)d * ED_TT + k], ea[(size_t)e * ED_TT + k]);
}
__global__ void k_loop_norm(float* __restrict__ la, const float* __restrict__ cnt, int N)
{
    int n = blockIdx.x * blockDim.x + threadIdx.x;
    if (n >= N) return;
    float inv = 1.f / fmaxf(cnt[n], 1.f);
    for (int k = 0; k < ED_TT; ++k) la[(size_t)n * ED_TT + k] *= inv;
}

// ---------------- edge phase ----------------
// pass 1: logits + segment max.  Edges e<Ereal use (src,dst,ea); e>=Ereal are
// self-loops (src=dst=e-Ereal, attr = la row) — only used for conv3.
__global__ void k_edge_logits(const int* __restrict__ src, const int* __restrict__ dst,
                              const float* __restrict__ ea, int edim,
                              const float* __restrict__ la, int Ereal, int Nloop,
                              const float* __restrict__ ssrc, const float* __restrict__ sdst,
                              const float* __restrict__ wev,
                              float* __restrict__ aw, unsigned* __restrict__ amax)
{
    int e = blockIdx.x * blockDim.x + threadIdx.x;
    if (e >= Ereal + Nloop) return;
    int s, d; const float* row;
    if (e < Ereal) { s = src[e]; d = dst[e]; row = ea + (size_t)e * edim; }
    else           { s = d = e - Ereal;       row = la + (size_t)s * edim; }
    #pragma unroll
    for (int h = 0; h < 2; ++h) {
        float t = ssrc[s * 2 + h] + sdst[d * 2 + h];
        for (int k = 0; k < edim; ++k) t += row[k] * wev[k * 2 + h];
        t = (t >= 0.f) ? t : 0.2f * t;              // leaky_relu(., 0.2)
        aw[(size_t)e * 2 + h] = t;
        atomicMax(&amax[d * 2 + h], encf(t));
    }
}
// pass 2: w = exp(a - amax[dst]); denom[dst] += w
__global__ void k_edge_exp(const int* __restrict__ dst, int Ereal, int Nloop,
                           float* __restrict__ aw, const unsigned* __restrict__ amax,
                           float* __restrict__ denom)
{
    int e = blockIdx.x * blockDim.x + threadIdx.x;
    if (e >= Ereal + Nloop) return;
    int d = (e < Ereal) ? dst[e] : (e - Ereal);
    #pragma unroll
    for (int h = 0; h < 2; ++h) {
        float w = __expf(aw[(size_t)e * 2 + h] - decf(amax[d * 2 + h]));
        aw[(size_t)e * 2 + h] = w;
        atomicAdd(&denom[d * 2 + h], w);
    }
}
// pass 3: outacc[dst, c] += hs[src, c] * w/(denom+1e-16)   — one thread per (edge, channel)
__global__ void k_edge_scatter(const int* __restrict__ src, const int* __restrict__ dst,
                               int Ereal, int Nloop,
                               const float* __restrict__ aw, const float* __restrict__ denom,
                               const float* __restrict__ hs, float* __restrict__ outacc)
{
    long long idx = (long long)blockIdx.x * blockDim.x + threadIdx.x;
    long long tot = (long long)(Ereal + Nloop) * HC;
    if (idx >= tot) return;
    int e = (int)(idx >> 7);
    int c = (int)(idx & 127);
    int h = c >> 6;
    int s, d;
    if (e < Ereal) { s = src[e]; d = dst[e]; } else { s = d = e - Ereal; }
    float coeff = aw[(size_t)e * 2 + h] / (denom[d * 2 + h] + 1e-16f);
    atomicAdd(&outacc[(size_t)d * HC + c], hs[(size_t)s * HC + c] * coeff);
}

// ---------------- finalize: head-mean + residual + bias, LN, leaky(0.01) ----------------
// one wave per node; lane handles channels l and l+32
__global__ void k_finalize(const float* __restrict__ outacc, const float* __restrict__ res,
                           int resLd, int resOff,
                           const float* __restrict__ bias,
                           const float* __restrict__ g, const float* __restrict__ beta,
                           float* __restrict__ dout, int outOff, int N)
{
    int wave = blockIdx.x * (blockDim.x >> 5) + ((int)threadIdx.x >> 5);
    int lane = threadIdx.x & 31;
    if (wave >= N) return;
    const float* acc = outacc + (size_t)wave * HC;
    const float* rr  = res + (size_t)wave * resLd + resOff;
    int c0 = lane, c1 = lane + 32;
    float v0 = 0.5f * (acc[c0] + acc[64 + c0]) + rr[c0] + bias[c0];
    float v1 = 0.5f * (acc[c1] + acc[64 + c1]) + rr[c1] + bias[c1];
    float mean = wred32(v0 + v1) * (1.f / 64.f);
    float d0 = v0 - mean, d1 = v1 - mean;
    float var = wred32(d0 * d0 + d1 * d1) * (1.f / 64.f);
    float inv = rsqrtf(var + 1e-5f);
    float y0 = d0 * inv * g[c0] + beta[c0];
    float y1 = d1 * inv * g[c1] + beta[c1];
    y0 = (y0 >= 0.f) ? y0 : 0.01f * y0;
    y1 = (y1 >= 0.f) ? y1 : 0.01f * y1;
    float* o = dout + (size_t)wave * OUTW + outOff;
    o[c0] = y0; o[c1] = y1;
}

__global__ void k_copy_tasks(const float* __restrict__ tx, float* __restrict__ dout)
{
    long long i = (long long)blockIdx.x * blockDim.x + threadIdx.x;
    if (i >= (long long)N_TASKS * D_T) return;
    int n = (int)(i >> 5);     // D_T == 32
    int k = (int)(i & 31);
    dout[(size_t)n * OUTW + k] = tx[i];
}

// =====================================================================
extern "C" void kernel_launch(void* const* d_in, const int* in_sizes, int n_in,
                              void* d_out, int out_size, void* d_ws, size_t ws_size,
                              hipStream_t stream) {
    const float* tasks_x   = (const float*)d_in[0];
    const float* data_x    = (const float*)d_in[1];
    const float* devices_x = (const float*)d_in[2];
    const float* ea_dt     = (const float*)d_in[3];
    const float* ea_dev    = (const float*)d_in[4];
    const float* ea_tt     = (const float*)d_in[5];

    // ---- resolve the flattened params (sorted-key pytree vs insertion order) ----
    const int P = 6;
    const float *Wd1,*We1,*Wres1,*Ws1,*ad1,*ae1,*as1,*b1;
    const float *Wd2,*We2,*Wres2,*Ws2,*ad2,*ae2,*as2,*b2;
    const float *W3 ,*We3,*Wres3,*ad3,*ae3,*as3,*b3;
    const float *ln1g,*ln1b,*ln2g,*ln2b,*ln4g,*ln4b;
    #define F(i) ((const float*)d_in[(i)])
    if (in_sizes[P] == D_T * HC) {
        // sorted keys: c1{Wd,We,Wres,Ws,ad,ae,as,b} c2{...} c3{W,We,Wres,ad,ae,as,b} ln1_b,ln1_g,...
        Wd1=F(P+0);  We1=F(P+1);  Wres1=F(P+2);  Ws1=F(P+3);  ad1=F(P+4);  ae1=F(P+5);  as1=F(P+6);  b1=F(P+7);
        Wd2=F(P+8);  We2=F(P+9);  Wres2=F(P+10); Ws2=F(P+11); ad2=F(P+12); ae2=F(P+13); as2=F(P+14); b2=F(P+15);
        W3 =F(P+16); We3=F(P+17); Wres3=F(P+18); ad3=F(P+19); ae3=F(P+20); as3=F(P+21); b3=F(P+22);
        ln1b=F(P+23); ln1g=F(P+24); ln2b=F(P+25); ln2g=F(P+26); ln4b=F(P+27); ln4g=F(P+28);
    } else {
        // insertion order: c1{We,as,ad,ae,b,Wres,Ws,Wd} c2{...} c3{We,as,ad,ae,b,Wres,W} ln1_g,ln1_b,...
        We1=F(P+0);  as1=F(P+1);  ad1=F(P+2);  ae1=F(P+3);  b1=F(P+4);  Wres1=F(P+5); Ws1=F(P+6); Wd1=F(P+7);
        We2=F(P+8);  as2=F(P+9);  ad2=F(P+10); ae2=F(P+11); b2=F(P+12); Wres2=F(P+13);Ws2=F(P+14);Wd2=F(P+15);
        We3=F(P+16); as3=F(P+17); ad3=F(P+18); ae3=F(P+19); b3=F(P+20); Wres3=F(P+21); W3=F(P+22);
        ln1g=F(P+23); ln1b=F(P+24); ln2g=F(P+25); ln2b=F(P+26); ln4g=F(P+27); ln4b=F(P+28);
    }
    #undef F
    const int EB = P + 29;
    const int* src_dt  = (const int*)d_in[EB + 0];
    const int* dst_dt  = (const int*)d_in[EB + 1];
    const int* src_dev = (const int*)d_in[EB + 2];
    const int* dst_dev = (const int*)d_in[EB + 3];
    const int* src_tt  = (const int*)d_in[EB + 4];
    const int* dst_tt  = (const int*)d_in[EB + 5];

    // ---- workspace carve-up (floats) ----
    float* w = (float*)d_ws;
    float* hs1   = w; w += (size_t)N_DATA  * HC;       // source projections
    float* hs3   = w; w += (size_t)N_TASKS * HC;
    float* hs2   = w; w += (size_t)N_DEVN  * HC;
    float* res   = w; w += (size_t)N_TASKS * 192;      // [res1|res2|res3]
    float* ssrc1 = w; w += (size_t)N_DATA  * 2;
    float* ssrc3 = w; w += (size_t)N_TASKS * 2;
    float* ssrc2 = w; w += (size_t)N_DEVN  * 2;
    float* sdst  = w; w += (size_t)N_TASKS * 2;        // reused per conv
    float* aw    = w; w += (size_t)E_DT    * 2;        // reused per conv (max E)
    unsigned* amax = (unsigned*)w; w += (size_t)N_TASKS * 2;
    float* denom = w; w += (size_t)N_TASKS * 2;
    float* outacc= w; w += (size_t)N_TASKS * HC;
    float* lcnt  = w; w += (size_t)N_TASKS;
    float* lattr = w; w += (size_t)N_TASKS * ED_TT;
    float* wdv   = w; w += 64;                         // D_dst<=32 * 2
    float* wev   = w; w += 64;

    auto nb = [](long long n, int b) { return (unsigned)((n + b - 1) / b); };
    const unsigned NEG_INF_ENC = 0x007FFFFFu;          // encf(-inf)

    // ---- dense projections (WMMA) ----
    // hs1 = data_x @ Ws1 : 1250*8 tiles
    k_gemm_wmma<<<nb((long long)(N_DATA/16)*(HC/16), 8), 256, 0, stream>>>(
        data_x, Ws1, hs1, N_DATA, D_D, HC, HC, 0);
    // hs3 = tasks_x @ W3 : 3125*8 tiles
    k_gemm_wmma<<<nb((long long)(N_TASKS/16)*(HC/16), 8), 256, 0, stream>>>(
        tasks_x, W3, hs3, N_TASKS, D_T, HC, HC, 0);
    // residuals into res[:, 0:64 | 64:128 | 128:192]
    k_gemm_wmma<<<nb((long long)(N_TASKS/16)*(CDIM/16), 8), 256, 0, stream>>>(
        tasks_x, Wres1, res, N_TASKS, D_T, CDIM, 192, 0);
    k_gemm_wmma<<<nb((long long)(N_TASKS/16)*(CDIM/16), 8), 256, 0, stream>>>(
        tasks_x, Wres2, res, N_TASKS, D_T, CDIM, 192, 64);
    k_gemm_wmma<<<nb((long long)(N_TASKS/16)*(CDIM/16), 8), 256, 0, stream>>>(
        tasks_x, Wres3, res, N_TASKS, D_T, CDIM, 192, 128);
    // hs2 = devices_x @ Ws2 (tiny)
    k_small_gemm<<<nb(N_DEVN * HC, 256), 256, 0, stream>>>(devices_x, Ws2, hs2, N_DEVN, D_V, HC);

    // per-node attention scalars from hs
    k_s_src<<<nb(N_DATA, 8), 256, 0, stream>>>(hs1, as1, ssrc1, N_DATA);
    k_s_src<<<nb(N_TASKS, 8), 256, 0, stream>>>(hs3, as3, ssrc3, N_TASKS);
    k_s_src<<<1, 256, 0, stream>>>(hs2, as2, ssrc2, N_DEVN);

    // self-loop attrs for conv3
    k_fill_f32<<<nb(N_TASKS, 256), 256, 0, stream>>>(lcnt, 0.f, N_TASKS);
    k_fill_f32<<<nb((long long)N_TASKS * ED_TT, 256), 256, 0, stream>>>(lattr, 0.f, (long long)N_TASKS * ED_TT);
    k_loop_stats<<<nb(E_TT, 256), 256, 0, stream>>>(dst_tt, ea_tt, lcnt, lattr, E_TT);
    k_loop_norm<<<nb(N_TASKS, 256), 256, 0, stream>>>(lattr, lcnt, N_TASKS);

    struct Conv {
        const int *src, *dst; const float* ea; int edim; int Ereal; int Nloop;
        const float *Wd, *ad, *We, *ae, *bias, *g, *beta, *ssrc, *hs; int Ddst; int col;
    };
    const Conv convs[3] = {
        { src_dt,  dst_dt,  ea_dt,  ED_DT,  E_DT,  0,       Wd1, ad1, We1, ae1, b1, ln1g, ln1b, ssrc1, hs1, D_T,  96 }, // d1
        { src_dev, dst_dev, ea_dev, ED_DEV, E_DEV, 0,       Wd2, ad2, We2, ae2, b2, ln2g, ln2b, ssrc2, hs2, D_T, 160 }, // d2
        { src_tt,  dst_tt,  ea_tt,  ED_TT,  E_TT,  N_TASKS, W3,  ad3, We3, ae3, b3, ln4g, ln4b, ssrc3, hs3, D_T,  32 }, // d3
    };
    const int resOffs[3] = { 0, 64, 128 };

    for (int ci = 0; ci < 3; ++ci) {
        const Conv& cv = convs[ci];
        const int Etot = cv.Ereal + cv.Nloop;
        // attention projection vectors + per-dst scalar
        k_vec_prep<<<1, 128, 0, stream>>>(cv.Wd, cv.ad, cv.We, cv.ae, wdv, wev, cv.Ddst, cv.edim);
        k_s_dst<<<nb(N_TASKS, 256), 256, 0, stream>>>(tasks_x, wdv, sdst, N_TASKS, cv.Ddst);
        // init segment buffers
        k_fill_u32<<<nb((long long)N_TASKS * 2, 256), 256, 0, stream>>>(amax, NEG_INF_ENC, (long long)N_TASKS * 2);
        k_fill_f32<<<nb((long long)N_TASKS * 2, 256), 256, 0, stream>>>(denom, 0.f, (long long)N_TASKS * 2);
        k_fill_f32<<<nb((long long)N_TASKS * HC, 256), 256, 0, stream>>>(outacc, 0.f, (long long)N_TASKS * HC);
        // edge passes
        k_edge_logits<<<nb(Etot, 256), 256, 0, stream>>>(cv.src, cv.dst, cv.ea, cv.edim,
                                                         lattr, cv.Ereal, cv.Nloop,
                                                         cv.ssrc, sdst, wev, aw, amax);
        k_edge_exp<<<nb(Etot, 256), 256, 0, stream>>>(cv.dst, cv.Ereal, cv.Nloop, aw, amax, denom);
        k_edge_scatter<<<nb((long long)Etot * HC, 256), 256, 0, stream>>>(cv.src, cv.dst,
                                                                          cv.Ereal, cv.Nloop,
                                                                          aw, denom, cv.hs, outacc);
        // head-mean + residual + bias + LN + leaky(0.01) -> d_out columns
        k_finalize<<<nb(N_TASKS, 8), 256, 0, stream>>>(outacc, res, 192, resOffs[ci],
                                                       cv.bias, cv.g, cv.beta,
                                                       (float*)d_out, cv.col, N_TASKS);
    }
    // raw task features -> cols [0,32)
    k_copy_tasks<<<nb((long long)N_TASKS * D_T, 256), 256, 0, stream>>>(tasks_x, (float*)d_out);

    (void)in_sizes; (void)n_in; (void)out_size; (void)ws_size;
}